// DCA3D_72739566125565
// MI455X (gfx1250) — compile-verified
//
#include <hip/hip_runtime.h>
#include <stdint.h>

// ---------------- problem geometry ----------------
#define Bn 4
#define Cn 16
#define Dn 96
#define Hn 96
#define Wn 96
#define SP (Dn * Hn * Wn)            // 884736 voxels per (b,c) channel
#define NCH (Bn * Cn)                // 64 channels
// tile: ZT x YT x 96 outputs per block
#define ZT 4
#define YT 8
#define TILES_Z (Dn / ZT)            // 24
#define TILES_Y (Hn / YT)            // 12
#define TILES_PER_CH (TILES_Z * TILES_Y)  // 288
#define NBLK (NCH * TILES_PER_CH)         // 18432
// LDS tile with halo: 6 planes x 10 rows x 104 floats (data at x-index 4..99)
#define ZP 6
#define YP 10
#define XP 104
#define LDS_FLOATS (ZP * YP * XP)    // 6240
#define NLOAD4 (ZP * YP * (Wn / 4))  // 1440 float4 chunks
#define NTHREADS 256
#define SEGW 12                      // x-elements per thread (8 segs * 12 = 96)

typedef float v4f __attribute__((ext_vector_type(4)));

// ---------------- CDNA5 async global->LDS path ----------------
__device__ __forceinline__ void async_load_b128(unsigned lds_byte_off,
                                                unsigned glb_byte_off,
                                                const float* sbase) {
  // VDST = per-lane LDS byte address, VADDR = per-lane 32-bit offset, SADDR = 64-bit base
  asm volatile("global_load_async_to_lds_b128 %0, %1, %2"
               :
               : "v"(lds_byte_off), "v"(glb_byte_off), "s"(sbase)
               : "memory");
}

__device__ __forceinline__ void wait_async0() {
#if __has_builtin(__builtin_amdgcn_s_wait_asynccnt)
  __builtin_amdgcn_s_wait_asynccnt(0);
#else
  asm volatile("s_wait_asynccnt 0" ::: "memory");
#endif
}

// Stage one halo tile (z0-1..z0+ZT, y0-1..y0+YT, full x rows, zero halos) into LDS.
__device__ __forceinline__ void load_tile(float* tile, const float* chbase,
                                          int z0, int y0, int tid) {
  // zero the x-halo columns (x = -1 -> idx 3, x = 96 -> idx 100)
  for (int l = tid; l < ZP * YP * 2; l += NTHREADS) {
    int side = l & 1;
    int row  = l >> 1;               // 0..59
    int ly = row % YP, lz = row / YP;
    tile[(lz * YP + ly) * XP + (side ? 100 : 3)] = 0.0f;
  }
  // interior rows: async b128 per lane (24 float4 per row), OOB rows zero-filled
  for (int l = tid; l < NLOAD4; l += NTHREADS) {
    int lx4 = l % 24;
    int row = l / 24;                // 0..59
    int ly = row % YP, lz = row / YP;
    int gz = z0 - 1 + lz;
    int gy = y0 - 1 + ly;
    int fidx = (lz * YP + ly) * XP + 4 + lx4 * 4;   // 16B aligned
    if ((unsigned)gz < (unsigned)Dn && (unsigned)gy < (unsigned)Hn) {
      unsigned ldsb = (unsigned)(uintptr_t)(&tile[fidx]);   // low 32b == LDS offset
      unsigned goff = (unsigned)(((gz * Hn + gy) * Wn + lx4 * 4) * 4);
      async_load_b128(ldsb, goff, chbase);
    } else {
      v4f z4 = (v4f)0.0f;
      *(v4f*)(&tile[fidx]) = z4;
    }
  }
  wait_async0();
  __syncthreads();
}

// Sliding-window 3x3x3 box stencil over a contiguous 12-wide x segment.
// Produces xv[i] (raw center) and tv[i] = x + alpha*(x - mean27).
__device__ __forceinline__ void compute_seg(const float* tile, int oz, int oy, int x0,
                                            float alpha, float* xv, float* tv) {
  const float* r[9];
#pragma unroll
  for (int dz = 0; dz < 3; ++dz)
#pragma unroll
    for (int dy = 0; dy < 3; ++dy)
      r[dz * 3 + dy] = tile + ((oz + dz) * YP + (oy + dy)) * XP + 4;

  float centc = 0.f, centn = 0.f;
  float cm = 0.f, cc = 0.f;
#pragma unroll
  for (int j = 0; j < 9; ++j) cm += r[j][x0 - 1];        // column x0-1 (halo-safe)
#pragma unroll
  for (int j = 0; j < 9; ++j) {
    float v = r[j][x0];
    cc += v;
    if (j == 4) centc = v;                                // dz=1,dy=1 center row
  }
#pragma unroll
  for (int i = 0; i < SEGW; ++i) {
    float cn = 0.f;
#pragma unroll
    for (int j = 0; j < 9; ++j) {
      float v = r[j][x0 + i + 1];
      cn += v;
      if (j == 4) centn = v;
    }
    float s  = cm + cc + cn;                              // 27-point sum
    float lm = s * (1.0f / 27.0f);
    float xc = centc;
    xv[i] = xc;
    tv[i] = xc + alpha * (xc - lm);
    cm = cc; cc = cn; centc = centn;
  }
}

// deterministic wave32 + cross-wave block reduction
__device__ __forceinline__ void block_reduce4(float& vmax, float& vmin,
                                              float& vsum, float& vsq,
                                              float red[4][8], int tid) {
#pragma unroll
  for (int off = 16; off > 0; off >>= 1) {
    vmax = fmaxf(vmax, __shfl_xor(vmax, off));
    vmin = fminf(vmin, __shfl_xor(vmin, off));
    vsum += __shfl_xor(vsum, off);
    vsq  += __shfl_xor(vsq,  off);
  }
  int wv = tid >> 5;
  if ((tid & 31) == 0) {
    red[0][wv] = vmax; red[1][wv] = vmin; red[2][wv] = vsum; red[3][wv] = vsq;
  }
  __syncthreads();
  if (tid == 0) {
#pragma unroll
    for (int i = 1; i < 8; ++i) {
      red[0][0] = fmaxf(red[0][0], red[0][i]);
      red[1][0] = fminf(red[1][0], red[1][i]);
      red[2][0] += red[2][i];
      red[3][0] += red[3][i];
    }
  }
  __syncthreads();
}

// ---------------- pass 1: per-tile stats of (x, t) ----------------
__global__ __launch_bounds__(NTHREADS) void dca_pass1(
    const float* __restrict__ x, const float* __restrict__ alpha,
    float* __restrict__ partials) {
  __shared__ __align__(16) float tile[LDS_FLOATS];
  __shared__ float red[4][8];
  const int blk = blockIdx.x;
  const int ch  = blk / TILES_PER_CH;
  const int tl  = blk % TILES_PER_CH;
  const int z0  = (tl / TILES_Y) * ZT;
  const int y0  = (tl % TILES_Y) * YT;
  const int tid = threadIdx.x;
  const float* chbase = x + (size_t)ch * SP;
  const float a = alpha[ch & (Cn - 1)];

  load_tile(tile, chbase, z0, y0, tid);

  // thread -> (seg, oy, oz): pure shifts
  const int seg = tid & 7;
  const int oy  = (tid >> 3) & 7;
  const int oz  = tid >> 6;

  float xv[SEGW], tv[SEGW];
  compute_seg(tile, oz, oy, seg * SEGW, a, xv, tv);

  float vmax = -3.402823466e38f, vmin = 3.402823466e38f, vsum = 0.f, vsq = 0.f;
#pragma unroll
  for (int i = 0; i < SEGW; ++i) {
    vmax = fmaxf(vmax, xv[i]);
    vmin = fminf(vmin, xv[i]);
    vsum += tv[i];
    vsq  += tv[i] * tv[i];
  }
  block_reduce4(vmax, vmin, vsum, vsq, red, tid);
  if (tid == 0) {
    float* p = partials + (size_t)blk * 4;
    p[0] = red[0][0]; p[1] = red[1][0]; p[2] = red[2][0]; p[3] = red[3][0];
  }
}

// ---------------- finalize: fold partials -> per-channel (A, B) ----------------
__global__ __launch_bounds__(NTHREADS) void dca_finalize(
    const float* __restrict__ partials, const float* __restrict__ gamma,
    const float* __restrict__ nw, const float* __restrict__ nb,
    float* __restrict__ finals) {
  __shared__ float red[4][8];
  const int ch  = blockIdx.x;
  const int tid = threadIdx.x;
  float vmax = -3.402823466e38f, vmin = 3.402823466e38f, vsum = 0.f, vsq = 0.f;
  for (int i = tid; i < TILES_PER_CH; i += NTHREADS) {
    const float* p = partials + ((size_t)ch * TILES_PER_CH + i) * 4;
    vmax = fmaxf(vmax, p[0]);
    vmin = fminf(vmin, p[1]);
    vsum += p[2];
    vsq  += p[3];
  }
  block_reduce4(vmax, vmin, vsum, vsq, red, tid);
  if (tid == 0) {
    const int c = ch & (Cn - 1);
    float range  = red[0][0] - red[1][0] + 1e-6f;     // gmax - gmin + EPS_DCA
    float scale  = gamma[c] * range;
    float mean_t = red[2][0] * (1.0f / (float)SP);
    float var_t  = fmaxf(red[3][0] * (1.0f / (float)SP) - mean_t * mean_t, 0.f);
    // out = scale*t + beta; beta cancels through instance-norm mean subtraction
    float rstd = rsqrtf(scale * scale * var_t + 1e-5f);  // EPS_IN
    float A  = scale * rstd * nw[c];
    float Bc = nb[c] - A * mean_t;
    finals[ch * 2 + 0] = A;
    finals[ch * 2 + 1] = Bc;
  }
}

// ---------------- pass 2: recompute t, apply affine + hswish ----------------
__global__ __launch_bounds__(NTHREADS) void dca_pass2(
    const float* __restrict__ x, const float* __restrict__ alpha,
    const float* __restrict__ finals, float* __restrict__ out) {
  __shared__ __align__(16) float tile[LDS_FLOATS];
  // Reverse block order: consume the tail of pass-1's stream (still L2-resident) first.
  const int blk = (NBLK - 1) - blockIdx.x;
  const int ch  = blk / TILES_PER_CH;
  const int tl  = blk % TILES_PER_CH;
  const int z0  = (tl / TILES_Y) * ZT;
  const int y0  = (tl % TILES_Y) * YT;
  const int tid = threadIdx.x;
  const float* chbase = x + (size_t)ch * SP;
  const float a  = alpha[ch & (Cn - 1)];
  const float A  = finals[ch * 2 + 0];
  const float Bc = finals[ch * 2 + 1];

  load_tile(tile, chbase, z0, y0, tid);

  const int seg = tid & 7;
  const int oy  = (tid >> 3) & 7;
  const int oz  = tid >> 6;

  float xv[SEGW], tv[SEGW];
  compute_seg(tile, oz, oy, seg * SEGW, a, xv, tv);

  float o[SEGW];
#pragma unroll
  for (int i = 0; i < SEGW; ++i) {
    float y = fmaf(A, tv[i], Bc);
    o[i] = y * fminf(fmaxf(y + 3.0f, 0.0f), 6.0f) * (1.0f / 6.0f);
  }

  // 3 aligned float4 non-temporal stores per thread (48B-aligned segment)
  float* orow = out + (size_t)ch * SP +
                ((size_t)(z0 + oz) * Hn + (y0 + oy)) * Wn + seg * SEGW;
#pragma unroll
  for (int q = 0; q < 3; ++q) {
    v4f v = { o[q * 4 + 0], o[q * 4 + 1], o[q * 4 + 2], o[q * 4 + 3] };
    __builtin_nontemporal_store(v, (v4f*)orow + q);
  }
}

// ---------------- host entry ----------------
extern "C" void kernel_launch(void* const* d_in, const int* in_sizes, int n_in,
                              void* d_out, int out_size, void* d_ws, size_t ws_size,
                              hipStream_t stream) {
  (void)in_sizes; (void)n_in; (void)out_size; (void)ws_size;
  const float* x     = (const float*)d_in[0];
  const float* alpha = (const float*)d_in[1];
  // const float* beta = (const float*)d_in[2];  // cancels analytically in instance norm
  const float* gamma = (const float*)d_in[3];
  const float* nw    = (const float*)d_in[4];
  const float* nb    = (const float*)d_in[5];
  float* out = (float*)d_out;

  float* partials = (float*)d_ws;                    // NBLK * 4 floats = 288 KiB
  float* finals   = partials + (size_t)NBLK * 4;     // NCH * 2 floats

  dca_pass1   <<<NBLK, NTHREADS, 0, stream>>>(x, alpha, partials);
  dca_finalize<<<NCH,  NTHREADS, 0, stream>>>(partials, gamma, nw, nb, finals);
  dca_pass2   <<<NBLK, NTHREADS, 0, stream>>>(x, alpha, finals, out);
}